// LCGLoss_72009421685001
// MI455X (gfx1250) — compile-verified
//
#include <hip/hip_runtime.h>
#include <math.h>

// CDNA5 / gfx1250. fp32 WMMA path: V_WMMA_F32_16X16X4_F32 (wave32) +
// async memory->LDS copies (GLOBAL_LOAD_ASYNC_TO_LDS_B128 / ASYNCcnt).
typedef __attribute__((ext_vector_type(2))) float v2f;
typedef __attribute__((ext_vector_type(8))) float v8f;
typedef __attribute__((ext_vector_type(4))) int   v4i;

#define Bc 256
#define Lc 512
#define Dc 512
#define MB 64                       // rows of (b,l) per block in the main pass
#define NPART ((Bc * Lc) / MB)      // 2048 column-partials
#define ASTR 516                    // padded LDS stride (floats): conflict-free b64 lanes
#define SSTR 260                    // padded LDS stride for S tile
#define SMEM_MAIN ((MB * ASTR + MB * SSTR) * sizeof(float))   // ~194 KB < 320 KB/WGP

#define NEG_INF (-INFINITY)

// async 16B memory->LDS copy (tracked by ASYNCcnt)
__device__ __forceinline__ void async_copy16(const float* g, float* l) {
    __builtin_amdgcn_global_load_async_to_lds_b128(
        (__attribute__((address_space(1))) v4i*)g,
        (__attribute__((address_space(3))) v4i*)l, 0, 0);
}

// streaming log-sum-exp helpers -----------------------------------------
__device__ __forceinline__ void lse_merge(float& m, float& s, float m2, float s2) {
    if (s2 <= 0.0f) return;                       // empty partial
    if (m2 > m) { s = s * __expf(m - m2) + s2; m = m2; }
    else        { s = s + s2 * __expf(m2 - m); }
}
__device__ __forceinline__ void lse_add(float& m, float& s, float x) {
    if (x > m) { s = s * __expf(m - x) + 1.0f; m = x; }
    else       { s += __expf(x - m); }
}

// WMMA wrapper: D = A(16x4) * B(4x16) + C ------------------------------
__device__ __forceinline__ v8f wmma4(v2f a, v2f b, v8f c) {
    return __builtin_amdgcn_wmma_f32_16x16x4_f32(false, a, false, b, (short)0, c,
                                                 false, false);
}

// ----------------------------------------------------------------------
__global__ void k_init(float* acc) {
    acc[0] = 0.0f;   // loss sum
    acc[1] = 0.0f;   // valid count
}

// ----------------------------------------------------------------------
// Kernel 1: image_proj[m,e] = sum_d image_features[m,d] * W_MV[e,d]
__global__ void k_img_proj(const float* __restrict__ imgf,
                           const float* __restrict__ W_MV,
                           float* __restrict__ img_proj) {
    int lane = threadIdx.x;                 // one wave32 per 16x16 tile
    int m0 = blockIdx.x * 16;
    int e0 = blockIdx.y * 16;
    int n  = lane & 15;
    int ko = (lane >> 4) * 2;
    int hf = lane >> 4;

    v8f acc = {};
    for (int kb = 0; kb < Dc; kb += 4) {
        v2f a, bb;
        const float* ap = imgf + (size_t)(m0 + n) * Dc + kb + ko;
        a.x = ap[0]; a.y = ap[1];
        const float* bp = W_MV + (size_t)(e0 + n) * Dc + kb + ko;  // B[d][e]=W_MV[e][d]
        bb.x = bp[0]; bb.y = bp[1];
        acc = wmma4(a, bb, acc);
    }
    for (int v = 0; v < 8; ++v)
        img_proj[(size_t)(m0 + hf * 8 + v) * Dc + e0 + n] = acc[v];
}

// ----------------------------------------------------------------------
// Kernel 2: M2[k,d] = sum_e image_proj[k,e] * W_ML[e,d]
// Folds both projections: S = text @ M2^T  (associativity of the chain).
__global__ void k_m2(const float* __restrict__ img_proj,
                     const float* __restrict__ W_ML,
                     float* __restrict__ M2) {
    int lane = threadIdx.x;
    int k0 = blockIdx.x * 16;               // 256 image rows
    int d0 = blockIdx.y * 16;               // 512 cols
    int n  = lane & 15;
    int ko = (lane >> 4) * 2;
    int hf = lane >> 4;

    v8f acc = {};
    for (int eb = 0; eb < Dc; eb += 4) {
        v2f a, bb;
        const float* ap = img_proj + (size_t)(k0 + n) * Dc + eb + ko;
        a.x = ap[0]; a.y = ap[1];
        bb.x = W_ML[(size_t)(eb + ko) * Dc + d0 + n];      // B[e][d]=W_ML[e][d]
        bb.y = W_ML[(size_t)(eb + ko + 1) * Dc + d0 + n];
        acc = wmma4(a, bb, acc);
    }
    for (int v = 0; v < 8; ++v)
        M2[(size_t)(k0 + hf * 8 + v) * Dc + d0 + n] = acc[v];
}

// ----------------------------------------------------------------------
// Kernel 3 (fused main pass), one block per 64-row tile of (b,l):
//   text tile (64x512) -> LDS via ASYNC b128 copies, double-buffered by
//   256-column halves against the WMMA k-loop
//   S = (text_tile x M2^T)/tau   (64x256, fp32 WMMA, LDS only)
//   row-wise masked LSE -> denom_log ; diagonal -> s_iji
//   per-column streaming (max,sum) partials -> part[]
__global__ void k_main(const float* __restrict__ text,
                       const float* __restrict__ M2,
                       const unsigned char* __restrict__ pad,   // jnp bool = 1 byte
                       const float* __restrict__ tau_p,
                       float* __restrict__ denom_ws,
                       float* __restrict__ siji_ws,
                       float2* __restrict__ part) {
    extern __shared__ float smem[];
    float* Abuf = smem;                 // MB x ASTR
    float* Sbuf = smem + MB * ASTR;     // MB x SSTR
    __shared__ int rv[MB];              // per-row validity of this tile

    const int tid  = threadIdx.x;       // 256 threads = 8 waves
    const int row0 = blockIdx.x * MB;   // flattened (b,l) row
    const int b    = row0 / Lc;         // tile never crosses batches (512%64==0)
    const int l0   = row0 % Lc;

    float tau = tau_p[0];
    if (tau < 1e-3f) tau = 1e-3f;
    const float itau = 1.0f / tau;

    // --- issue async memory->LDS copies for BOTH 256-col halves up front.
    // Each wave issues 16 b128 ops per half; async loads complete in-order
    // per wave, so asynccnt<=16 means "my half-0 writes landed".
    for (int h = 0; h < 2; ++h) {
#pragma unroll
        for (int it = 0; it < 16; ++it) {
            int j = tid + 256 * it;          // float4 index within this half
            int r = j >> 6, c4 = j & 63;     // 64 float4 per half-row
            const float* g = text + (size_t)(row0 + r) * Dc + h * 256 + c4 * 4;
            float* l = Abuf + r * ASTR + h * 256 + c4 * 4;
            async_copy16(g, l);
        }
    }
    if (tid < MB) rv[tid] = pad[(size_t)b * Lc + l0 + tid] ? 0 : 1;

    const int wave = tid >> 5;
    const int lane = tid & 31;
    const int n  = lane & 15;
    const int ko = (lane >> 4) * 2;
    const int hf = lane >> 4;

    // accumulators for both k-tiles owned by this wave, live across halves
    v8f acc[2][4] = {{{}, {}, {}, {}}, {{}, {}, {}, {}}};

    for (int h = 0; h < 2; ++h) {
        if (h == 0) __builtin_amdgcn_s_wait_asynccnt(16);  // half-0 landed
        else        __builtin_amdgcn_s_wait_asynccnt(0);   // half-1 landed
        __syncthreads();                                   // block-wide visibility

#pragma unroll
        for (int kk = 0; kk < 2; ++kk) {
            int k0 = (wave + kk * 8) * 16;
            for (int db = h * 256; db < h * 256 + 256; db += 4) {
                v2f bb;
                const float* bp = M2 + (size_t)(k0 + n) * Dc + db + ko; // B[d][k]=M2[k][d]
                bb.x = bp[0]; bb.y = bp[1];
#pragma unroll
                for (int mt = 0; mt < 4; ++mt) {
                    v2f a;
                    const float* apl = Abuf + (mt * 16 + n) * ASTR + db + ko;
                    a.x = apl[0]; a.y = apl[1];
                    acc[kk][mt] = wmma4(a, bb, acc[kk][mt]);
                }
            }
        }
    }

#pragma unroll
    for (int kk = 0; kk < 2; ++kk) {
        int k0 = (wave + kk * 8) * 16;
        for (int mt = 0; mt < 4; ++mt)
            for (int v = 0; v < 8; ++v)
                Sbuf[(mt * 16 + hf * 8 + v) * SSTR + k0 + n] = acc[kk][mt][v] * itau;
    }
    __syncthreads();

    // denom_log[row] = LSE_k{ valid[k,l] } S[row,k] ; 4 lanes per row
    {
        int row = tid >> 2;          // 0..63
        int sub = tid & 3;           // 64 columns each
        int l   = l0 + row;
        float m = NEG_INF, s = 0.0f;
        for (int k = sub * 64; k < sub * 64 + 64; ++k)
            if (!pad[(size_t)k * Lc + l]) lse_add(m, s, Sbuf[row * SSTR + k]);
        for (int off = 1; off < 4; off <<= 1) {   // width-4 xor tree inside wave32
            float m2 = __shfl_xor(m, off, 4);
            float s2 = __shfl_xor(s, off, 4);
            lse_merge(m, s, m2, s2);
        }
        if (sub == 0) {
            denom_ws[row0 + row] = (s > 0.0f) ? (m + __logf(s)) : 0.0f;
            siji_ws[row0 + row]  = Sbuf[row * SSTR + b];   // diagonal entry s_iji
        }
    }

    // per-column streaming-LSE partial over this tile (mask: row valid AND k != b)
    {
        int k = tid;                 // blockDim.x == Bc == 256
        float m = NEG_INF, s = 0.0f;
        if (k != b) {
            for (int r = 0; r < MB; ++r)
                if (rv[r]) lse_add(m, s, Sbuf[r * SSTR + k]);
        }
        part[(size_t)blockIdx.x * Bc + k] = make_float2(m, s);
    }
}

// ----------------------------------------------------------------------
// Kernel 4: fold 2048 (max,sum) partials per column -> cross_log[k]
__global__ void k_cross(const float2* __restrict__ part,
                        float* __restrict__ cross_log) {
    __shared__ float sm[256], ss[256];
    int k = blockIdx.x, tid = threadIdx.x;
    float m = NEG_INF, s = 0.0f;
    for (int i = tid; i < NPART; i += blockDim.x) {
        float2 p = part[(size_t)i * Bc + k];
        lse_merge(m, s, p.x, p.y);
    }
    sm[tid] = m; ss[tid] = s;
    __syncthreads();
    for (int off = 128; off > 0; off >>= 1) {
        if (tid < off) {
            float mm = sm[tid], sv = ss[tid];
            lse_merge(mm, sv, sm[tid + off], ss[tid + off]);
            sm[tid] = mm; ss[tid] = sv;
        }
        __syncthreads();
    }
    if (tid == 0)
        cross_log[k] = (ss[0] > 0.0f) ? (sm[0] + __logf(ss[0])) : NEG_INF;
}

// ----------------------------------------------------------------------
// Kernel 5: per-row loss terms, masked sum + count into accumulators
__global__ void k_loss(const float* __restrict__ denom_ws,
                       const float* __restrict__ siji_ws,
                       const float* __restrict__ cross_log,
                       const unsigned char* __restrict__ pad,
                       float* __restrict__ acc) {
    __shared__ float sl[256], sc[256];
    int tid = threadIdx.x;
    int row = blockIdx.x * blockDim.x + tid;       // 0..131071 == b*L + l
    float contrib = 0.0f, cnt = 0.0f;
    if (!pad[row]) {
        int bb = row >> 9;                         // L == 512
        float sii = siji_ws[row];
        float dl  = denom_ws[row];
        float cl  = cross_log[bb];
        float mx  = fmaxf(sii, cl);
        float neg = mx + __logf(__expf(sii - mx) + __expf(cl - mx)); // logaddexp
        contrib = 0.5f * (dl + neg) - sii;         // 0.5*((dl-sii)+(neg-sii))
        cnt = 1.0f;
    }
    sl[tid] = contrib; sc[tid] = cnt;
    __syncthreads();
    for (int off = 128; off > 0; off >>= 1) {
        if (tid < off) { sl[tid] += sl[tid + off]; sc[tid] += sc[tid + off]; }
        __syncthreads();
    }
    if (tid == 0) {
        atomicAdd(&acc[0], sl[0]);
        atomicAdd(&acc[1], sc[0]);
    }
}

// ----------------------------------------------------------------------
__global__ void k_fin(const float* __restrict__ acc, float* __restrict__ out) {
    out[0] = acc[0] / acc[1];
}

// ----------------------------------------------------------------------
extern "C" void kernel_launch(void* const* d_in, const int* in_sizes, int n_in,
                              void* d_out, int out_size, void* d_ws, size_t ws_size,
                              hipStream_t stream) {
    (void)in_sizes; (void)n_in; (void)out_size; (void)ws_size;
    const float* text  = (const float*)d_in[0];          // (B,L,D)
    const float* imgf  = (const float*)d_in[1];          // (B,D)
    const float* W_ML  = (const float*)d_in[2];          // (D,D)
    const float* W_MV  = (const float*)d_in[3];          // (D,D)
    const float* tau   = (const float*)d_in[4];          // (1,)
    const unsigned char* pad = (const unsigned char*)d_in[5]; // (B,L) bool

    float* ws = (float*)d_ws;
    const size_t OFF_IMG  = 0;                            // img_proj 256*512
    const size_t OFF_M2   = OFF_IMG + (size_t)Bc * Dc;    // M2       256*512
    const size_t OFF_DEN  = OFF_M2  + (size_t)Bc * Dc;    // denom    131072
    const size_t OFF_SIJ  = OFF_DEN + (size_t)Bc * Lc;    // s_iji    131072
    const size_t OFF_CRO  = OFF_SIJ + (size_t)Bc * Lc;    // cross    256
    const size_t OFF_ACC  = OFF_CRO + Bc;                 // accum    8
    const size_t OFF_PART = OFF_ACC + 8;                  // 2048*256 float2 (~4 MB)
    float*  img_proj  = ws + OFF_IMG;
    float*  M2        = ws + OFF_M2;
    float*  denom_ws  = ws + OFF_DEN;
    float*  siji_ws   = ws + OFF_SIJ;
    float*  cross_log = ws + OFF_CRO;
    float*  acc       = ws + OFF_ACC;
    float2* part      = (float2*)(ws + OFF_PART);

    // main kernel needs ~194 KB dynamic LDS (>64 KB default cap)
    (void)hipFuncSetAttribute((const void*)k_main,
                              hipFuncAttributeMaxDynamicSharedMemorySize,
                              (int)SMEM_MAIN);

    k_init<<<1, 1, 0, stream>>>(acc);
    k_img_proj<<<dim3(Bc / 16, Dc / 16), 32, 0, stream>>>(imgf, W_MV, img_proj);
    k_m2<<<dim3(Bc / 16, Dc / 16), 32, 0, stream>>>(img_proj, W_ML, M2);
    k_main<<<(Bc * Lc) / MB, 256, SMEM_MAIN, stream>>>(
        text, M2, pad, tau, denom_ws, siji_ws, part);
    k_cross<<<Bc, 256, 0, stream>>>(part, cross_log);
    k_loss<<<(Bc * Lc) / 256, 256, 0, stream>>>(denom_ws, siji_ws, cross_log, pad, acc);
    k_fin<<<1, 1, 0, stream>>>(acc, (float*)d_out);
}